// HBV_33758442946665
// MI455X (gfx1250) — compile-verified
//
#include <hip/hip_runtime.h>
#include <stdint.h>

// ---------------------------------------------------------------------------
// HBV hydrological model scan for MI455X (gfx1250).
//
// Memory-bound problem (~70 MB effective traffic -> ~3 us at 23.3 TB/s; the
// 58 MB dynamic-parameter stream also fits in the 192 MB L2). Only parameter
// indices 0 (BETA) and 12 (BETAET) are time-varying; the other 11 are frozen
// at t = staind. The two dynamic rows are streamed with the CDNA5 Tensor
// Data Mover (async DMA -> LDS, tracked by TENSORcnt), triple-buffered and
// prefetched two 8-step chunks ahead, because the serial scan + one
// workgroup per WGP cannot hide memory latency any other way.
// ---------------------------------------------------------------------------

typedef unsigned int u32x4 __attribute__((ext_vector_type(4)));
typedef int          i32x8 __attribute__((ext_vector_type(8)));
typedef int          i32x4 __attribute__((ext_vector_type(4)));

#define NSTEP  365
#define NGRID  1000
#define MU     16
#define NPAR   13
#define GPB    16                 // grid cells per block
#define TCHUNK 8                  // timesteps per TDM chunk
#define NBUF   3                  // LDS buffers (prefetch depth 2)
#define NCHUNK ((NSTEP + TCHUNK - 1) / TCHUNK)   // 46

// Issue one TDM load: tile = [chunkSteps][ngb grids][2 params {0,12}][16 mu]
// from parameters[t0 .. t0+cs, g0 .. g0+ngb, {0,12}, 0..15] into LDS (packed).
__device__ __forceinline__ void tdm_issue_chunk(const float* params, int t0, int cs,
                                                int g0, int ngb, uint32_t ldsAddr) {
    uint64_t ga = (uint64_t)(uintptr_t)(params +
                    ((uint64_t)t0 * NGRID + g0) * (NPAR * MU));
    // ---- D# group 0 (128b): count=1, lds_addr, global_addr[56:0], type=2
    u32x4 g0v;
    g0v[0] = 1u;                                               // count=1, user mode
    g0v[1] = ldsAddr;                                          // LDS byte address
    g0v[2] = (uint32_t)ga;                                     // global_addr[31:0]
    g0v[3] = ((uint32_t)(ga >> 32) & 0x01FFFFFFu) | (2u << 30);// [56:32] | type=2
    // ---- D# group 1 (256b)
    i32x8 g1v;
    g1v[0] = (int)(2u << 16);                  // workgroup_mask=0, data_size=2 (4B)
    g1v[1] = (int)((uint32_t)MU << 16);        // tensor_dim0[15:0]=16
    g1v[2] = (int)(2u << 16);                  // tensor_dim0 hi=0 | tensor_dim1=2
    g1v[3] = (int)((uint32_t)MU << 16);        // tensor_dim1 hi=0 | tile_dim0=16
    g1v[4] = (int)(2u | ((uint32_t)ngb << 16));// tile_dim1=2 | tile_dim2=ngb
    g1v[5] = 12 * MU;                          // tensor_dim0_stride = 192 (param 0 -> 12)
    g1v[6] = (int)((uint32_t)(NPAR * MU) << 16); // d0str hi=0 | dim1_stride lo=208 (grid)
    g1v[7] = 0;                                // dim1_stride hi
    // ---- D# group 2 (128b): dim2 = grids, dim3 = timesteps
    i32x4 g2v;
    g2v[0] = ngb;                              // tensor_dim2
    g2v[1] = cs;                               // tensor_dim3
    g2v[2] = NGRID * NPAR * MU;                // tensor_dim2_stride = 208000 (timestep)
    g2v[3] = (int)((uint32_t)cs << 16);        // stride hi=0 | tile_dim3 = cs
    // ---- D# group 3: dim4 unused
    i32x4 g3v = {0, 0, 0, 0};
    // amdgpu-toolchain (clang-23) 6-arg form: extra int32x8 (zero-filled) + cpol.
    i32x8 g4v = {0, 0, 0, 0, 0, 0, 0, 0};
    __builtin_amdgcn_tensor_load_to_lds(g0v, g1v, g2v, g3v, g4v, 0);
}

__global__ __launch_bounds__(256)
void hbv_scan_kernel(const float* __restrict__ x,
                     const float* __restrict__ params,
                     const int*   __restrict__ staindPtr,
                     float*       __restrict__ out) {
    // triple-buffered dynamic-parameter tiles: [buf][step][grid][param2][mu]
    __shared__ float dynBuf[NBUF][TCHUNK * GPB * 2 * MU];   // 3 x 16 KB

    const int tid = threadIdx.x;
    const int m   = tid & (MU - 1);       // ensemble member (16 contiguous lanes)
    const int gl  = tid >> 4;             // local grid cell 0..15
    const int g0  = blockIdx.x * GPB;
    const int ngb = min(GPB, NGRID - g0);
    const int gle = min(gl, ngb - 1);     // clamped for tail block
    const int g   = g0 + gle;
    const bool active = (gl < ngb);
    const bool wv0    = (tid >> 5) == 0;  // wave 0 drives the TDM
    const int staind  = *staindPtr;

    const float lo[NPAR] = {1.0f, 50.0f, 0.05f, 0.01f, 0.001f, 0.2f, 0.0f,
                            0.0f, -2.5f, 0.5f, 0.0f, 0.0f, 0.3f};
    const float hi[NPAR] = {6.0f, 1000.0f, 0.9f, 0.5f, 0.2f, 1.0f, 10.0f,
                            100.0f, 2.5f, 10.0f, 0.1f, 0.2f, 5.0f};

    // Static parameters (indices 1..11) frozen at t = staind. Coalesced: the
    // 16 mu lanes read 64 contiguous bytes per parameter.
    const float* pst = params + ((uint64_t)staind * NGRID + g) * (NPAR * MU) + m;
    const float parFC    = lo[1]  + pst[1  * MU] * (hi[1]  - lo[1]);
    const float parK0    = lo[2]  + pst[2  * MU] * (hi[2]  - lo[2]);
    const float parK1    = lo[3]  + pst[3  * MU] * (hi[3]  - lo[3]);
    const float parK2    = lo[4]  + pst[4  * MU] * (hi[4]  - lo[4]);
    const float parLP    = lo[5]  + pst[5  * MU] * (hi[5]  - lo[5]);
    const float parPERC  = lo[6]  + pst[6  * MU] * (hi[6]  - lo[6]);
    const float parUZL   = lo[7]  + pst[7  * MU] * (hi[7]  - lo[7]);
    const float parTT    = lo[8]  + pst[8  * MU] * (hi[8]  - lo[8]);
    const float parCFMAX = lo[9]  + pst[9  * MU] * (hi[9]  - lo[9]);
    const float parCFR   = lo[10] + pst[10 * MU] * (hi[10] - lo[10]);
    const float parCWH   = lo[11] + pst[11 * MU] * (hi[11] - lo[11]);

    float SNOWPACK = 1e-3f, MELTWATER = 1e-3f, SM = 1e-3f, SUZ = 1e-3f, SLZ = 1e-3f;

    // Prologue: put two chunks in flight (EXEC-independent, wave-level DMA).
    if (wv0) {
        tdm_issue_chunk(params, 0, TCHUNK, g0, ngb,
                        (uint32_t)(uintptr_t)&dynBuf[0][0]);
        tdm_issue_chunk(params, TCHUNK, TCHUNK, g0, ngb,
                        (uint32_t)(uintptr_t)&dynBuf[1][0]);
    }

    int t = 0;
    int bufIdx = 0;
    int issueBuf = 2;                       // next buffer slot to fill
    for (int c = 0; c < NCHUNK; ++c) {
        const int cs = min(TCHUNK, NSTEP - c * TCHUNK);
        if (wv0) {
            if (c + 2 < NCHUNK) {
                const int t2 = (c + 2) * TCHUNK;
                // Prefetch chunk c+2 into buf[(c+2)%3] (safe: trailing barrier
                // of iteration c-1 released it), then wait for chunk c only:
                // three chunks are in flight and TDM completes in order.
                tdm_issue_chunk(params, t2, min(TCHUNK, NSTEP - t2), g0, ngb,
                                (uint32_t)(uintptr_t)&dynBuf[issueBuf][0]);
                __builtin_amdgcn_s_wait_tensorcnt(2);
            } else if (c + 1 < NCHUNK) {
                __builtin_amdgcn_s_wait_tensorcnt(1);
            } else {
                __builtin_amdgcn_s_wait_tensorcnt(0);
            }
        }
        __syncthreads();                    // publish buffer c to all waves
        const float* buf = dynBuf[bufIdx];

        for (int s = 0; s < cs; ++s, ++t) {
            const int bi = ((s * ngb + gle) * 2) * MU + m;
            const float parBETA   = lo[0]  + buf[bi]      * (hi[0]  - lo[0]);
            const float parBETAET = lo[12] + buf[bi + MU] * (hi[12] - lo[12]);

            const float* xc = x + ((uint64_t)t * NGRID + g) * 3;
            const float Pt = xc[0], Tt = xc[1], Et = xc[2];

            // ---- HBV step (matches reference order exactly) ----
            const float RAIN = (Tt >= parTT) ? Pt : 0.0f;
            const float SNOW = (Tt <  parTT) ? Pt : 0.0f;
            SNOWPACK += SNOW;
            const float melt = fminf(fmaxf(parCFMAX * (Tt - parTT), 0.0f), SNOWPACK);
            MELTWATER += melt;
            SNOWPACK  -= melt;
            const float refreezing =
                fminf(fmaxf(parCFR * parCFMAX * (parTT - Tt), 0.0f), MELTWATER);
            SNOWPACK  += refreezing;
            MELTWATER -= refreezing;
            const float tosoil = fmaxf(MELTWATER - parCWH * SNOWPACK, 0.0f);
            MELTWATER -= tosoil;
            float sw = powf(SM / parFC, parBETA);
            sw = fminf(fmaxf(sw, 0.0f), 1.0f);
            const float recharge = (RAIN + tosoil) * sw;
            SM += RAIN + tosoil - recharge;
            const float excess = fmaxf(SM - parFC, 0.0f);
            SM -= excess;
            float ef = powf(SM / (parLP * parFC), parBETAET);
            ef = fminf(fmaxf(ef, 0.0f), 1.0f);
            const float ETact = fminf(SM, Et * ef);
            SM = fmaxf(SM - ETact, 1e-5f);
            SUZ += recharge + excess;
            const float PERC = fminf(SUZ, parPERC);
            SUZ -= PERC;
            const float Q0 = parK0 * fmaxf(SUZ - parUZL, 0.0f);
            SUZ -= Q0;
            const float Q1 = parK1 * SUZ;
            SUZ -= Q1;
            SLZ += PERC;
            const float Q2 = parK2 * SLZ;
            SLZ -= Q2;

            // ---- ensemble means over the 16 mu lanes (wave32, width-16 xor) ----
            float s0 = Q0, s1 = Q1, s2 = Q2, s3 = ETact;
            #pragma unroll
            for (int off = 8; off > 0; off >>= 1) {
                s0 += __shfl_xor(s0, off, 16);
                s1 += __shfl_xor(s1, off, 16);
                s2 += __shfl_xor(s2, off, 16);
                s3 += __shfl_xor(s3, off, 16);
            }
            if (m == 0 && active) {
                float* o = out + ((uint64_t)t * NGRID + g) * 5;
                const float inv = 1.0f / (float)MU;
                o[0] = (s0 + s1 + s2) * inv;   // Qsimave
                o[1] = s0 * inv;               // Q0 mean
                o[2] = s1 * inv;               // Q1 mean
                o[3] = s2 * inv;               // Q2 mean
                o[4] = s3 * inv;               // ET mean
            }
        }
        __syncthreads();                    // done reading buf before it is reused
        bufIdx   = (bufIdx   + 1 == NBUF) ? 0 : bufIdx + 1;
        issueBuf = (issueBuf + 1 == NBUF) ? 0 : issueBuf + 1;
    }
}

extern "C" void kernel_launch(void* const* d_in, const int* in_sizes, int n_in,
                              void* d_out, int out_size, void* d_ws, size_t ws_size,
                              hipStream_t stream) {
    const float* x      = (const float*)d_in[0];   // (365,1000,3)
    const float* params = (const float*)d_in[1];   // (365,1000,13,16)
    // d_in[2] = rtwts  (unused by the reference output)
    const int*   staind = (const int*)d_in[3];     // scalar
    // d_in[4] = mu (structural constant 16, baked in)
    float* out = (float*)d_out;                    // (365,1000,5)

    dim3 grid((NGRID + GPB - 1) / GPB);            // 63 blocks
    hipLaunchKernelGGL(hbv_scan_kernel, grid, dim3(256), 0, stream,
                       x, params, staind, out);
}